// GCNModel_15728170238728
// MI455X (gfx1250) — compile-verified
//
#include <hip/hip_runtime.h>

typedef __attribute__((ext_vector_type(2))) float v2f;
typedef __attribute__((ext_vector_type(4))) float v4f;
typedef __attribute__((ext_vector_type(8))) float v8f;

#define TILE_ROWS 128  // 8 waves * 16 rows

// -------------------------------------------------------------------------
// Dense per-relation transform: XW[r] = H[j(r)] @ W[r]
//   H : [2, N, din]   W : [4, din, 64]   XW : [4, N, 64]
// One wave computes a 16x64 tile with V_WMMA_F32_16X16X4_F32 (full f32).
// W is staged in LDS interleaved by K-pairs so every B fragment is one
// aligned ds_load_b64: wlds[(k/2)*128 + col*2 + (k&1)] = W[k][col].
// -------------------------------------------------------------------------
__global__ __launch_bounds__(256) void gemm_wmma_f32(
    const float* __restrict__ H, const float* __restrict__ W,
    float* __restrict__ XW, int N, int din)
{
    __shared__ float wlds[128 * 64];   // up to 32 KB

    const int r   = blockIdx.y;        // relation 0..3
    const int j   = r & 1;             // source node-type per relation
    const int tid = threadIdx.x;

    const float* Wr = W + (size_t)r * din * 64;
    for (int i = tid; i < din * 64; i += 256) {
        int k = i >> 6, col = i & 63;
        wlds[(k >> 1) * 128 + col * 2 + (k & 1)] = Wr[i];
    }
    __syncthreads();

    const int wave = tid >> 5;
    const int lane = tid & 31;
    const int ln15 = lane & 15;
    const int hi   = lane >> 4;        // 0: lanes 0-15, 1: lanes 16-31
    const int m0   = blockIdx.x * TILE_ROWS + wave * 16;

    // A-matrix 16x4 f32 layout: lane l<16 -> row m0+l, K = k..k+1
    //                           lane l>=16 -> row m0+l-16, K = k+2..k+3
    int arow = m0 + ln15;
    if (arow >= N) arow = N - 1;       // clamp (stores guarded); EXEC stays full
    const float* hrow = H + (size_t)j * N * din + (size_t)arow * din;
    const int koff = hi * 2;

    v8f c0 = {}, c1 = {}, c2 = {}, c3 = {};

    for (int k = 0; k < din; k += 4) {
        v2f a = *(const v2f*)(hrow + k + koff);                 // global_load_b64
        const float* wp = wlds + ((k + koff) >> 1) * 128;       // K-pair row
        v2f b0 = *(const v2f*)(wp + (ln15     ) * 2);           // ds_load_b64
        v2f b1 = *(const v2f*)(wp + (ln15 + 16) * 2);
        v2f b2 = *(const v2f*)(wp + (ln15 + 32) * 2);
        v2f b3 = *(const v2f*)(wp + (ln15 + 48) * 2);
        c0 = __builtin_amdgcn_wmma_f32_16x16x4_f32(false, a, false, b0, (short)0, c0, false, false);
        c1 = __builtin_amdgcn_wmma_f32_16x16x4_f32(false, a, false, b1, (short)0, c1, false, false);
        c2 = __builtin_amdgcn_wmma_f32_16x16x4_f32(false, a, false, b2, (short)0, c2, false, false);
        c3 = __builtin_amdgcn_wmma_f32_16x16x4_f32(false, a, false, b3, (short)0, c3, false, false);
    }

    // C/D layout: VGPR p -> row m0+p (lanes 0-15) / m0+p+8 (lanes 16-31), col = lane%16
    float* xwr = XW + (size_t)r * N * 64;
#pragma unroll
    for (int p = 0; p < 8; ++p) {
        int row = m0 + p + hi * 8;
        if (row < N) {
            float* o = xwr + (size_t)row * 64 + ln15;
            o[ 0] = c0[p];
            o[16] = c1[p];
            o[32] = c2[p];
            o[48] = c3[p];
        }
    }
}

// -------------------------------------------------------------------------
// Sparse scatter: out[r>>1][rows[e]] += vals[e] * XW[r][cols[e]]
// One edge per 16-lane half-wave; float4 per lane (256B row gather, L2-resident).
// -------------------------------------------------------------------------
__global__ __launch_bounds__(256) void spmm_scatter(
    const int* __restrict__ rows, const int* __restrict__ cols,
    const float* __restrict__ vals, const float* __restrict__ XW,
    float* __restrict__ out, int N, int E)
{
    const int r    = blockIdx.y;            // relation 0..3
    const int tid  = threadIdx.x;
    const int half = tid >> 4;              // 16 edges per 256-thread block
    const int ln   = tid & 15;

    long e = (long)blockIdx.x * 16 + half;
    if (e >= E) return;

    const long base = (long)r * E + e;
    const int   row = rows[base];
    const int   col = cols[base];
    const float v   = vals[base];

    const float* src = XW + ((size_t)r * N + col) * 64 + ln * 4;
    v4f x = *(const v4f*)src;               // global_load_b128

    float* dst = out + ((size_t)(r >> 1) * N + row) * 64 + ln * 4;
    unsafeAtomicAdd(dst + 0, v * x.x);      // global_atomic_add_f32 (L2)
    unsafeAtomicAdd(dst + 1, v * x.y);
    unsafeAtomicAdd(dst + 2, v * x.z);
    unsafeAtomicAdd(dst + 3, v * x.w);
}

__global__ __launch_bounds__(256) void relu_inplace(float* __restrict__ x, long n4)
{
    long i = (long)blockIdx.x * blockDim.x + threadIdx.x;
    if (i < n4) {
        v4f v = ((v4f*)x)[i];
        v.x = fmaxf(v.x, 0.f);
        v.y = fmaxf(v.y, 0.f);
        v.z = fmaxf(v.z, 0.f);
        v.w = fmaxf(v.w, 0.f);
        ((v4f*)x)[i] = v;
    }
}

extern "C" void kernel_launch(void* const* d_in, const int* in_sizes, int n_in,
                              void* d_out, int out_size, void* d_ws, size_t ws_size,
                              hipStream_t stream)
{
    const float* feat = (const float*)d_in[0];
    const int*   a1r  = (const int*)d_in[1];
    const int*   a1c  = (const int*)d_in[2];
    const float* a1v  = (const float*)d_in[3];
    const int*   a2r  = (const int*)d_in[4];
    const int*   a2c  = (const int*)d_in[5];
    const float* a2v  = (const float*)d_in[6];
    const float* W1   = (const float*)d_in[7];
    const float* W2   = (const float*)d_in[8];
    const float* W3   = (const float*)d_in[9];
    const float* W4   = (const float*)d_in[10];

    const int F = in_sizes[7] / (4 * 64);    // 128
    const int N = in_sizes[0] / (2 * F);     // 50000
    const int E = in_sizes[1] / 4;           // 500000

    float* XW = (float*)d_ws;                         // [4, N, 64]  51.2 MB
    float* hA = XW + (size_t)4 * N * 64;              // [2, N, 64]  25.6 MB
    float* hB = hA + (size_t)2 * N * 64;              // [2, N, 64]  25.6 MB
    const size_t outBytes = (size_t)2 * N * 64 * sizeof(float);

    const dim3 blk(256);
    const dim3 gGemm((N + TILE_ROWS - 1) / TILE_ROWS, 4);
    const dim3 gSp((E + 15) / 16, 4);
    const long n4 = (long)2 * N * 64 / 4;
    const int  gRelu = (int)((n4 + 255) / 256);

    // Layer 1: feat (din=F) -> hA, adj1, ReLU
    gemm_wmma_f32<<<gGemm, blk, 0, stream>>>(feat, W1, XW, N, F);
    hipMemsetAsync(hA, 0, outBytes, stream);
    spmm_scatter<<<gSp, blk, 0, stream>>>(a1r, a1c, a1v, XW, hA, N, E);
    relu_inplace<<<gRelu, blk, 0, stream>>>(hA, n4);

    // Layer 2: hA (din=64) -> hB, adj1, ReLU
    gemm_wmma_f32<<<gGemm, blk, 0, stream>>>(hA, W2, XW, N, 64);
    hipMemsetAsync(hB, 0, outBytes, stream);
    spmm_scatter<<<gSp, blk, 0, stream>>>(a1r, a1c, a1v, XW, hB, N, E);
    relu_inplace<<<gRelu, blk, 0, stream>>>(hB, n4);

    // Layer 3: hB (din=64) -> hA, adj2, ReLU
    gemm_wmma_f32<<<gGemm, blk, 0, stream>>>(hB, W3, XW, N, 64);
    hipMemsetAsync(hA, 0, outBytes, stream);
    spmm_scatter<<<gSp, blk, 0, stream>>>(a2r, a2c, a2v, XW, hA, N, E);
    relu_inplace<<<gRelu, blk, 0, stream>>>(hA, n4);

    // Layer 4: hA (din=64) -> d_out, adj2, no ReLU
    gemm_wmma_f32<<<gGemm, blk, 0, stream>>>(hA, W4, XW, N, 64);
    hipMemsetAsync(d_out, 0, outBytes, stream);
    spmm_scatter<<<gSp, blk, 0, stream>>>(a2r, a2c, a2v, XW, (float*)d_out, N, E);
}